// QuantumGraphQuanvolutionHybrid_65481071400168
// MI455X (gfx1250) — compile-verified
//
#include <hip/hip_runtime.h>
#include <hip/hip_bf16.h>
#include <math.h>

typedef __attribute__((ext_vector_type(2))) float v2f;
typedef __attribute__((ext_vector_type(8))) float v8f;

// ---------------------------------------------------------------------------
// Kernel 1: quanvolution features via V_WMMA_F32_16X16X4_F32
//   One wave32 per tile of 16 consecutive (image,patch) pairs.
//   phi = psi @ U^T done as 4 K-chunks x {Re,Im} = 8 WMMAs per tile, fp32 exact.
//   Measurement sums via a per-wave LDS transpose (8 b32 stores + 4 b128 loads)
//   instead of 128 lane-permutes.
// ---------------------------------------------------------------------------
__global__ void __launch_bounds__(256)
quanv_wmma_kernel(const float* __restrict__ x,
                  const float* __restrict__ Ur,
                  const float* __restrict__ Ui,
                  float* __restrict__ feat,
                  int total_patches,   // B * 196
                  int n_tiles)
{
    __shared__ float lds_probs[8 * 256];            // 8 waves x (16 patches x 16 states)

    const int lane = threadIdx.x & 31;
    const int wave = blockIdx.x * (blockDim.x >> 5) + (threadIdx.x >> 5);
    if (wave >= n_tiles) return;                    // wave-uniform exit (no block barrier used)

    const int half = lane >> 4;                     // which 16-lane half
    const int nn   = lane & 15;                     // A row (patch-in-tile) AND B column (state n)
    const int gp0  = wave * 16;                     // first global patch of this tile

    // ---- B operands: B[k][n] = U^T[k][n] = U[n][k], split into 4 K-chunks ----
    // f32 B 4x16 layout: VGPR0 = row K=4k+2*half (N = lane&15), VGPR1 = next row.
    v2f br[4], bi[4];
#pragma unroll
    for (int k = 0; k < 4; ++k) {
        const int kc = 4 * k + 2 * half;
        br[k].x = Ur[nn * 16 + kc];
        br[k].y = Ur[nn * 16 + kc + 1];
        bi[k].x = Ui[nn * 16 + kc];
        bi[k].y = Ui[nn * 16 + kc + 1];
    }

    // ---- per-lane patch -> angles -> 4-qubit product state psi[16] ----
    int gp = gp0 + nn;
    if (gp >= total_patches) gp = total_patches - 1;  // clamp: EXEC must stay full for WMMA
    const int b  = gp / 196;
    const int p  = gp - b * 196;
    const int pr = p / 14;
    const int pc = p - pr * 14;
    const float* px = x + (size_t)b * 784 + (size_t)(2 * pr) * 28 + (size_t)(2 * pc);
    const float2 row0 = *(const float2*)(px);        // angles (0,0),(0,1)
    const float2 row1 = *(const float2*)(px + 28);   // angles (1,0),(1,1)

    float s0, c0, s1, c1, s2, c2, s3, c3;
    sincosf(row0.x * 0.5f, &s0, &c0);   // qubit 0
    sincosf(row0.y * 0.5f, &s1, &c1);   // qubit 1
    sincosf(row1.x * 0.5f, &s2, &c2);   // qubit 2
    sincosf(row1.y * 0.5f, &s3, &c3);   // qubit 3

    const float q0[2] = {c0, s0}, q1[2] = {c1, s1}, q2[2] = {c2, s2}, q3[2] = {c3, s3};
    float psi[16];
#pragma unroll
    for (int n = 0; n < 16; ++n)
        psi[n] = q0[(n >> 3) & 1] * q1[(n >> 2) & 1] * q2[(n >> 1) & 1] * q3[n & 1];

    // ---- phi = psi x U^T : 4 K-chunks, Re and Im accumulators ----
    v8f acc_re = {};
    v8f acc_im = {};
#pragma unroll
    for (int k = 0; k < 4; ++k) {
        v2f a;
        a.x = psi[4 * k + 2 * half];        // A f32 16x4 layout: VGPR0 = K=4k+2*half
        a.y = psi[4 * k + 2 * half + 1];    //                    VGPR1 = K+1
        acc_re = __builtin_amdgcn_wmma_f32_16x16x4_f32(
            false, a, false, br[k], (short)0, acc_re, false, false);
        acc_im = __builtin_amdgcn_wmma_f32_16x16x4_f32(
            false, a, false, bi[k], (short)0, acc_im, false, false);
    }

    // ---- |phi|^2 -> per-wave LDS tile (transpose D layout -> patch-major) ----
    // D layout: VGPR r holds patch (r + 8*half), state n = lane&15.
    float* myp = lds_probs + (threadIdx.x >> 5) * 256;
#pragma unroll
    for (int r = 0; r < 8; ++r) {
        const float re = acc_re[r];
        const float im = acc_im[r];
        myp[(r + 8 * half) * 16 + nn] = re * re + im * im;
    }

    // Same-wave LDS ops are in-order; wait for our stores and fence the compiler.
    asm volatile("s_wait_dscnt 0" ::: "memory");

    // ---- each lane gathers one patch's 16 probs (4 x ds_load_b128) ----
    const float4 v0 = *(const float4*)(myp + nn * 16 + 0);    // m = 0..3
    const float4 v1 = *(const float4*)(myp + nn * 16 + 4);    // m = 4..7
    const float4 v2 = *(const float4*)(myp + nn * 16 + 8);    // m = 8..11
    const float4 v3 = *(const float4*)(myp + nn * 16 + 12);   // m = 12..15

    // p0: bit3 set (m>=8); p1: bit2; p2: bit1; p3: bit0
    const float p0 = (v2.x + v2.y + v2.z + v2.w) + (v3.x + v3.y + v3.z + v3.w);
    const float p1 = (v1.x + v1.y + v1.z + v1.w) + (v3.x + v3.y + v3.z + v3.w);
    const float p2 = (v0.z + v0.w + v1.z + v1.w) + (v2.z + v2.w + v3.z + v3.w);
    const float p3 = (v0.y + v0.w + v1.y + v1.w) + (v2.y + v2.w + v3.y + v3.w);

    const int patch = gp0 + nn;
    if (half == 0 && patch < total_patches) {
        *(float4*)(feat + (size_t)patch * 4) = make_float4(p0, p1, p2, p3);
    }
}

// ---------------------------------------------------------------------------
// Kernel 2: logits = feat @ W^T + b, then log_softmax. One wave per row.
// W is 10x784 = 31KB -> L2 resident; feat streamed once.
// ---------------------------------------------------------------------------
__global__ void __launch_bounds__(256)
classifier_kernel(const float* __restrict__ feat,
                  const float* __restrict__ W,
                  const float* __restrict__ bias,
                  float* __restrict__ out,
                  int rows)
{
    const int lane = threadIdx.x & 31;
    const int row  = blockIdx.x * (blockDim.x >> 5) + (threadIdx.x >> 5);
    if (row >= rows) return;

    const float* f = feat + (size_t)row * 784;
    float acc[10];
#pragma unroll
    for (int c = 0; c < 10; ++c) acc[c] = 0.0f;

    for (int k = lane; k < 784; k += 32) {
        const float v = f[k];
#pragma unroll
        for (int c = 0; c < 10; ++c) acc[c] += v * W[c * 784 + k];
    }
#pragma unroll
    for (int c = 0; c < 10; ++c) {
#pragma unroll
        for (int d = 16; d >= 1; d >>= 1) acc[c] += __shfl_xor(acc[c], d, 32);
    }

    float lg[10];
    float mx = -3.402823e38f;
#pragma unroll
    for (int c = 0; c < 10; ++c) {
        lg[c] = acc[c] + bias[c];
        mx = fmaxf(mx, lg[c]);
    }
    float se = 0.0f;
#pragma unroll
    for (int c = 0; c < 10; ++c) se += expf(lg[c] - mx);
    const float lse = mx + logf(se);

    if (lane < 10) out[(size_t)row * 10 + lane] = lg[lane] - lse;
}

// ---------------------------------------------------------------------------
extern "C" void kernel_launch(void* const* d_in, const int* in_sizes, int n_in,
                              void* d_out, int out_size, void* d_ws, size_t ws_size,
                              hipStream_t stream)
{
    const float* x    = (const float*)d_in[0];   // (B, 28, 28)
    const float* Ur   = (const float*)d_in[1];   // (16, 16)
    const float* Ui   = (const float*)d_in[2];   // (16, 16)
    const float* W    = (const float*)d_in[3];   // (10, 784)
    const float* bias = (const float*)d_in[4];   // (10,)
    float* out  = (float*)d_out;                 // (B, 10)
    float* feat = (float*)d_ws;                  // (B, 784) scratch

    const int Bsz           = in_sizes[0] / 784;
    const int total_patches = Bsz * 196;
    const int n_tiles       = (total_patches + 15) / 16;

    // 8 waves (256 threads) per block
    const int blocks1 = (n_tiles + 7) / 8;
    quanv_wmma_kernel<<<blocks1, 256, 0, stream>>>(x, Ur, Ui, feat,
                                                   total_patches, n_tiles);

    const int blocks2 = (Bsz + 7) / 8;           // one wave per row
    classifier_kernel<<<blocks2, 256, 0, stream>>>(feat, W, bias, out, Bsz);
}